// RelativeAttention_71476845740463
// MI455X (gfx1250) — compile-verified
//
#include <hip/hip_runtime.h>

typedef float v2f __attribute__((ext_vector_type(2)));
typedef float v8f __attribute__((ext_vector_type(8)));

// D = A(16x4,f32) * B(4x16,f32) + C(16x16,f32)   — CDNA5 V_WMMA_F32_16X16X4_F32
#define WMMA_F32(a, b, c) \
  __builtin_amdgcn_wmma_f32_16x16x4_f32(false, (a), false, (b), (short)0, (c), false, false)

constexpr int Bc = 2, Lc = 1024, Dc = 1024, Hc = 16, HDc = 64, Pc = 2047;
constexpr float SCALE = 0.125f; // 1/sqrt(64)

// ---------------------------------------------------------------------------
// Out[M,N] = X[M,K] @ W[K,N] + bias[N]
// fp32 WMMA; each wave owns a 64x32 block: 4 M-subtiles x 2 N-subtiles.
// Per K-chunk (4): 4 A-frag b64 loads + 4 B-frag b32 loads feed 8 WMMAs.
// ---------------------------------------------------------------------------
__global__ __launch_bounds__(256) void gemm_f32_wmma(
    const float* __restrict__ X, const float* __restrict__ W,
    const float* __restrict__ bias, float* __restrict__ Out,
    int M, int N, int K)
{
  const int lane = threadIdx.x & 31;
  const int wave = threadIdx.x >> 5;
  const int lo = lane & 15;
  const int hi = lane >> 4;
  const int nblk = N >> 5;                          // 32-wide N blocks
  const int tile = blockIdx.x * 8 + wave;           // wave-uniform
  if (tile >= (M >> 6) * nblk) return;              // uniform exit: EXEC all-1
  const int m0 = (tile / nblk) << 6;
  const int n0 = (tile % nblk) << 5;

  // A frag (subtile t): lane -> row m0+16t+lo, K = k + 2*hi + {0,1} (float2)
  // B frag (subtile u): lane -> col n0+16u+lo, K = k + 2*hi + {0,1} (2 strided)
  const float* xr0 = X + (size_t)(m0 + lo) * K;
  const float* xr1 = xr0 + (size_t)16 * K;
  const float* xr2 = xr1 + (size_t)16 * K;
  const float* xr3 = xr2 + (size_t)16 * K;
  const float* wp  = W + (size_t)(2 * hi) * N + n0 + lo;

  v8f acc[8];
#pragma unroll
  for (int i = 0; i < 8; ++i) acc[i] = (v8f){};

  for (int k = 0; k < K; k += 4) {
    const int ka = k + 2 * hi;
    v2f a0 = *(const v2f*)(xr0 + ka);
    v2f a1 = *(const v2f*)(xr1 + ka);
    v2f a2 = *(const v2f*)(xr2 + ka);
    v2f a3 = *(const v2f*)(xr3 + ka);
    v2f b0, b1;
    b0.x = wp[0];  b0.y = wp[N];
    b1.x = wp[16]; b1.y = wp[N + 16];
    acc[0] = WMMA_F32(a0, b0, acc[0]);
    acc[1] = WMMA_F32(a0, b1, acc[1]);
    acc[2] = WMMA_F32(a1, b0, acc[2]);
    acc[3] = WMMA_F32(a1, b1, acc[3]);
    acc[4] = WMMA_F32(a2, b0, acc[4]);
    acc[5] = WMMA_F32(a2, b1, acc[5]);
    acc[6] = WMMA_F32(a3, b0, acc[6]);
    acc[7] = WMMA_F32(a3, b1, acc[7]);
    wp += (size_t)4 * N;
  }

  const float bv0 = bias[n0 + lo];
  const float bv1 = bias[n0 + 16 + lo];
#pragma unroll
  for (int t = 0; t < 4; ++t) {
#pragma unroll
    for (int g = 0; g < 8; ++g) {                   // C layout: row +g+8*hi, col lo
      float* orow = Out + (size_t)(m0 + 16 * t + g + 8 * hi) * N + n0;
      orow[lo]      = acc[2 * t][g]     + bv0;
      orow[16 + lo] = acc[2 * t + 1][g] + bv1;
    }
  }
}

// ---------------------------------------------------------------------------
// Fused causal attention with relative-position bias (skew trick via WMMA).
// One wave per (b, h, 16-row i-tile).  qkv layout: [B, L, 3*D] (q|k|v).
// ctx layout: [B, L, D] with column h*HD+d  (the 'bihd' reshape).
// ---------------------------------------------------------------------------
__global__ __launch_bounds__(32) void attn_rel(
    const float* __restrict__ qkv, const float* __restrict__ Er,
    float* __restrict__ ctx)
{
  __shared__ float Sbuf[16][17];   // raw QK^T tile
  __shared__ float Gbuf[16][33];   // bias window tile (32 offsets)
  __shared__ float Pbuf[16][18];   // probabilities (stride 18 keeps 8B align)
  __shared__ float rowM[16];
  __shared__ float rowS[16];
  __shared__ float rowF[16];

  const int lane = threadIdx.x & 31;
  const int lo = lane & 15, hi = lane >> 4;
  const int it = blockIdx.x, h = blockIdx.y, b = blockIdx.z;
  const int i0 = it << 4;
  const float NEG_INF = -__builtin_inff();

  // Q fragments for the whole i-tile: 16 K-chunks of float2 per lane
  const float* qbase = qkv + (size_t)(b * Lc + i0 + lo) * (3 * Dc) + h * HDc;
  v2f qa[16];
#pragma unroll
  for (int c = 0; c < 16; ++c) qa[c] = *(const v2f*)(qbase + 4 * c + 2 * hi);

  v8f o0 = {}, o1 = {}, o2 = {}, o3 = {};          // O accum: 16x64
  if (hi == 0) { rowM[lo] = NEG_INF; rowS[lo] = 0.0f; }
  __syncthreads();

  const float* erh = Er + (size_t)h * Pc * HDc;

  for (int jt = 0; jt <= it; ++jt) {
    const int j0 = jt << 4;

    // ---- S = Q K^T  (16 WMMAs)
    const float* kbase = qkv + (size_t)(b * Lc + j0 + lo) * (3 * Dc) + Dc + h * HDc;
    v8f sacc = {};
#pragma unroll
    for (int c = 0; c < 16; ++c) {
      v2f kb = *(const v2f*)(kbase + 4 * c + 2 * hi);
      sacc = WMMA_F32(qa[c], kb, sacc);
    }

    // ---- G = Q Ewin^T : 32 offset columns d = (j0-i0-15) .. (j0-i0+16)
    //      Er row = (d % P) with python-style wrap (d >= -(L-1)-15 > -P here)
    const int dmin = j0 - i0 - 15;
    const int d0 = dmin + lo;       const int r0 = (d0 < 0) ? d0 + Pc : d0;
    const int d1 = dmin + 16 + lo;  const int r1 = (d1 < 0) ? d1 + Pc : d1;
    const float* e0 = erh + (size_t)r0 * HDc;
    const float* e1 = erh + (size_t)r1 * HDc;
    v8f g0 = {}, g1 = {};
#pragma unroll
    for (int c = 0; c < 16; ++c) {
      v2f eb0 = *(const v2f*)(e0 + 4 * c + 2 * hi);
      v2f eb1 = *(const v2f*)(e1 + 4 * c + 2 * hi);
      g0 = WMMA_F32(qa[c], eb0, g0);
      g1 = WMMA_F32(qa[c], eb1, g1);
    }

    // ---- stage tiles to LDS
#pragma unroll
    for (int g = 0; g < 8; ++g) {
      const int r = g + 8 * hi;
      Sbuf[r][lo]      = sacc[g];
      Gbuf[r][lo]      = g0[g];
      Gbuf[r][16 + lo] = g1[g];
    }
    __syncthreads();

    // ---- online softmax on 16 row lanes; bias read is the skewed gather
    if (hi == 0) {
      const int r = lo;
      const float mOld = rowM[r];
      float vals[16];
      float mt = NEG_INF;
#pragma unroll
      for (int c = 0; c < 16; ++c) {
        float v = (j0 + c <= i0 + r)
                    ? Sbuf[r][c] * SCALE + Gbuf[r][c - r + 15]
                    : NEG_INF;
        vals[c] = v;
        mt = fmaxf(mt, v);
      }
      const float mNew = fmaxf(mOld, mt);
      const float f = __expf(mOld - mNew);         // 0 on first tile
      float ss = 0.0f;
#pragma unroll
      for (int c = 0; c < 16; ++c) {
        float p = __expf(vals[c] - mNew);          // exp(-inf)=0 for masked
        Pbuf[r][c] = p;
        ss += p;
      }
      rowM[r] = mNew;
      rowS[r] = rowS[r] * f + ss;
      rowF[r] = f;
    }
    __syncthreads();

    // ---- rescale running O by per-row factor
    float fr[8];
#pragma unroll
    for (int g = 0; g < 8; ++g) fr[g] = rowF[g + 8 * hi];
#pragma unroll
    for (int g = 0; g < 8; ++g) {
      o0[g] *= fr[g]; o1[g] *= fr[g]; o2[g] *= fr[g]; o3[g] *= fr[g];
    }

    // ---- O += P V  (P re-laid-out as A-fragments via LDS; 16 WMMAs)
    const float* vbase = qkv + (size_t)(b * Lc + j0) * (3 * Dc) + 2 * Dc + h * HDc;
#pragma unroll
    for (int c = 0; c < 4; ++c) {
      v2f pa = *(const v2f*)(&Pbuf[lo][4 * c + 2 * hi]);
      const float* vr0 = vbase + (size_t)(4 * c + 2 * hi) * (3 * Dc);
      const float* vr1 = vr0 + 3 * Dc;
      v2f vb;
      vb.x = vr0[lo];      vb.y = vr1[lo];      o0 = WMMA_F32(pa, vb, o0);
      vb.x = vr0[16 + lo]; vb.y = vr1[16 + lo]; o1 = WMMA_F32(pa, vb, o1);
      vb.x = vr0[32 + lo]; vb.y = vr1[32 + lo]; o2 = WMMA_F32(pa, vb, o2);
      vb.x = vr0[48 + lo]; vb.y = vr1[48 + lo]; o3 = WMMA_F32(pa, vb, o3);
    }
    __syncthreads();
  }

  // ---- normalize and write ctx[b][i][h*HD + n]
  float inv[8];
#pragma unroll
  for (int g = 0; g < 8; ++g) inv[g] = 1.0f / rowS[g + 8 * hi];
  float* obase = ctx + (size_t)(b * Lc + i0) * Dc + h * HDc;
#pragma unroll
  for (int g = 0; g < 8; ++g) {
    const int r = g + 8 * hi;
    float* orow = obase + (size_t)r * Dc;
    orow[lo]      = o0[g] * inv[g];
    orow[16 + lo] = o1[g] * inv[g];
    orow[32 + lo] = o2[g] * inv[g];
    orow[48 + lo] = o3[g] * inv[g];
  }
}

// ---------------------------------------------------------------------------
extern "C" void kernel_launch(void* const* d_in, const int* in_sizes, int n_in,
                              void* d_out, int out_size, void* d_ws, size_t ws_size,
                              hipStream_t stream) {
  (void)in_sizes; (void)n_in; (void)out_size; (void)ws_size;
  const float* x     = (const float*)d_in[0];
  // d_in[1] is the boolean mask: causal triu(k=1), handled analytically.
  const float* W_qkv = (const float*)d_in[2];
  const float* b_qkv = (const float*)d_in[3];
  const float* W_o   = (const float*)d_in[4];
  const float* b_o   = (const float*)d_in[5];
  const float* Er    = (const float*)d_in[6];
  float* out = (float*)d_out;

  float* qkv = (float*)d_ws;                               // [B, L, 3D]  24 MB
  float* ctx = qkv + (size_t)Bc * Lc * 3 * Dc;             // [B, L, D]    8 MB

  const int M = Bc * Lc;

  {   // qkv = x @ W_qkv + b_qkv          (M=2048, N=3072, K=1024)
    int tiles = (M / 64) * ((3 * Dc) / 32);
    gemm_f32_wmma<<<(tiles + 7) / 8, 256, 0, stream>>>(x, W_qkv, b_qkv, qkv,
                                                       M, 3 * Dc, Dc);
  }
  {   // fused relative attention
    dim3 grid(Lc / 16, Hc, Bc);
    attn_rel<<<grid, 32, 0, stream>>>(qkv, Er, ctx);
  }
  {   // out = ctx @ W_o + b_o            (M=2048, N=1024, K=1024)
    int tiles = (M / 64) * (Dc / 32);
    gemm_f32_wmma<<<(tiles + 7) / 8, 256, 0, stream>>>(ctx, W_o, b_o, out,
                                                       M, Dc, Dc);
  }
}